// LearnableTopP_84516366451092
// MI455X (gfx1250) — compile-verified
//
#include <hip/hip_runtime.h>
#include <hip/hip_bf16.h>
#include <stdint.h>

// Problem: B=4, H=16, S=1024. One workgroup per (b,h,s) row of 1024 floats.
// Per row: TDM-load the row into LDS, pack (value_bits, ~index) 64-bit keys,
// bitonic-sort descending in LDS, write the first k indices. k comes from
// out_size on the host (the reference's data-dependent cutoff is a Python-side
// scalar, identical for all rows).
//
// Barrier schedule: thread t's compare-exchange index map keeps wave w
// (threads 32w..32w+31) inside key block [64w, 64w+64) for all strides <= 32.
// LDS ops from one wave are processed in order (CDNA5 ISA, DScnt), so those
// stages need only a compiler fence, not a workgroup barrier. Only the 10
// stages with stride >= 64 (sz >= 128) need s_barrier.

#define S_LEN    1024
#define NTHREADS 512   // 16 wave32 waves; one compare-exchange per thread/stage

typedef unsigned int v4u __attribute__((ext_vector_type(4)));
typedef int          v8i __attribute__((ext_vector_type(8)));
typedef int          v4i __attribute__((ext_vector_type(4)));

#ifndef __has_builtin
#define __has_builtin(x) 0
#endif

#if __has_builtin(__builtin_amdgcn_tensor_load_to_lds) && \
    __has_builtin(__builtin_amdgcn_s_wait_tensorcnt)
#define HAVE_TDM 1
#else
#define HAVE_TDM 0
#endif

// Zero-instruction compiler fence: stops clang from caching LDS values in
// VGPRs across wave-local stages. Hardware ordering is already guaranteed
// (same-wave LDS ops are in order).
#define WAVE_FENCE() asm volatile("" ::: "memory")

__device__ __forceinline__ void cmpex(unsigned long long* keys, unsigned t,
                                      unsigned sz, unsigned stride) {
  const unsigned i = ((t & ~(stride - 1u)) << 1) | (t & (stride - 1u));
  const unsigned j = i + stride;
  const unsigned long long a = keys[i];
  const unsigned long long b = keys[j];
  const bool desc   = ((i & sz) == 0u);
  const bool doswap = desc ? (a < b) : (a > b);
  if (doswap) { keys[i] = b; keys[j] = a; }
}

__global__ __launch_bounds__(NTHREADS)
void LearnableTopP_topk_kernel(const float* __restrict__ atn,
                               int* __restrict__ out, int k) {
  __shared__ float              stage[S_LEN];   // 4 KB raw row (TDM target)
  __shared__ unsigned long long keys[S_LEN];    // 8 KB sort keys

  const unsigned t   = threadIdx.x;
  const unsigned row = blockIdx.x;
  const float* rowp  = atn + (size_t)row * S_LEN;

#if HAVE_TDM
  // Wave 0 issues one TDM descriptor for the whole 4 KB row. TDM ignores EXEC,
  // so gate at wave granularity (compiler emits s_cbranch_execz around this).
  if (t < 32) {
    const uint64_t ga  = (uint64_t)(uintptr_t)rowp;
    const uint32_t lds = (uint32_t)(uintptr_t)&stage[0];
    // D# group 0: count=1 (valid user desc), lds_addr, global_addr[56:0], type=2
    v4u g0 = { 1u,
               lds,
               (uint32_t)(ga & 0xFFFFFFFFu),
               (uint32_t)(((ga >> 32) & 0x01FFFFFFu) | 0x80000000u) };
    // D# group 1 (bit-packed per ISA 8.4):
    //  w0: workgroup_mask=0, data_size=2 (4B), no pad/iterate/atomic-barrier
    //  w1[31:16]=tensor_dim0.lo16=1024 ; w2[31:16]=tensor_dim1.lo16=1
    //  w3[31:16]=tile_dim0=1024       ; w4[15:0]=tile_dim1=1, tile_dim2=0
    //  w5=tensor_dim0_stride.lo32=1024; w6,w7=0
    v8i g1 = { 0x00020000,
               0x04000000,
               0x00010000,
               0x04000000,
               0x00000001,
               1024, 0, 0 };
    v4i z4 = {0, 0, 0, 0};
#if __clang_major__ >= 23
    v8i z8 = {0, 0, 0, 0, 0, 0, 0, 0};
    __builtin_amdgcn_tensor_load_to_lds(g0, g1, z4, z4, z8, 0);
#else
    __builtin_amdgcn_tensor_load_to_lds(g0, g1, z4, z4, 0);
#endif
    __builtin_amdgcn_s_wait_tensorcnt(0);
  }
  __syncthreads();
#else
  stage[t]            = rowp[t];
  stage[t + NTHREADS] = rowp[t + NTHREADS];
  __syncthreads();
#endif

  // Pack keys: softmax values are positive, so the IEEE bit pattern is order-
  // preserving. Low word = ~index so ties break toward the smaller index
  // under a descending u64 sort (matches stable argsort of -atn).
  {
    const unsigned i0 = t, i1 = t + NTHREADS;
    keys[i0] = ((unsigned long long)__float_as_uint(stage[i0]) << 32) |
               (unsigned long long)(0xFFFFFFFFu - i0);
    keys[i1] = ((unsigned long long)__float_as_uint(stage[i1]) << 32) |
               (unsigned long long)(0xFFFFFFFFu - i1);
  }
  __syncthreads();

  // Phases sz=2..64: every stride <= 32 -> fully wave-local, zero barriers.
  for (unsigned sz = 2; sz <= 64; sz <<= 1) {
    for (unsigned stride = sz >> 1; stride > 0; stride >>= 1) {
      cmpex(keys, t, sz, stride);
      WAVE_FENCE();
    }
  }
  __syncthreads();

  // Phases sz=128..1024: strides >= 64 cross waves (barrier per stage);
  // strides <= 32 are wave-local again (fence only), one barrier per phase.
  for (unsigned sz = 128; sz <= S_LEN; sz <<= 1) {
    for (unsigned stride = sz >> 1; stride >= 64; stride >>= 1) {
      cmpex(keys, t, sz, stride);
      __syncthreads();
    }
    for (unsigned stride = 32; stride > 0; stride >>= 1) {
      cmpex(keys, t, sz, stride);
      WAVE_FENCE();
    }
    __syncthreads();
  }

  // Emit top-k indices (descending value order); index is the low DWORD of the
  // key, so read 32 bits instead of the full 64-bit key.
  const unsigned* klo = (const unsigned*)keys;  // little-endian low words at 2*o
  for (unsigned o = t; o < (unsigned)k; o += NTHREADS) {
    out[(size_t)row * (unsigned)k + o] = (int)(0xFFFFFFFFu - klo[2u * o]);
  }
}

extern "C" void kernel_launch(void* const* d_in, const int* in_sizes, int n_in,
                              void* d_out, int out_size, void* d_ws, size_t ws_size,
                              hipStream_t stream) {
  const float* atn = (const float*)d_in[0];   // (4,16,1024,1024) fp32
  // d_in[1] = threshold: its effect is already folded into out_size (k).
  const int rows = 4 * 16 * 1024;             // 65536
  int k = out_size / rows;
  if (k < 1) k = 1;
  if (k > S_LEN) k = S_LEN;
  LearnableTopP_topk_kernel<<<dim3(rows), dim3(NTHREADS), 0, stream>>>(
      atn, (int*)d_out, k);
}